// CPUForgetMult_48404281426014
// MI455X (gfx1250) — compile-verified
//
#include <hip/hip_runtime.h>
#include <hip/hip_bf16.h>
#include <stdint.h>

// ---------------------------------------------------------------------------
// QRNN forget-mult scan on MI455X (gfx1250).
// Memory-bound (536 MB @ 23.3 TB/s ~= 23us). Strategy: per-wave TDM
// (tensor_load_to_lds) double-buffered streaming of f/z/i tiles into LDS,
// serial fp32 FMA scan per channel, coalesced stores.
// ---------------------------------------------------------------------------

typedef unsigned int u32x4 __attribute__((ext_vector_type(4)));
typedef int          i32x4 __attribute__((ext_vector_type(4)));
typedef int          i32x8 __attribute__((ext_vector_type(8)));

#define CHUNK   16            // time steps per tile
#define TILE_W  32            // channels per wave (wave32: 1 channel/lane)
#define WAVES   8
#define BLOCK   (WAVES * 32)  // 256 threads

// Issue one TDM 2D tile load: CHUNK rows x TILE_W fp32, row stride = BH elems.
// D# built per CDNA5 ISA §8.3 (group0) / §8.4 (group1); groups 2-3 zero (2D).
__device__ __forceinline__ void tdm_load_tile(const float* gsrc,
                                              uint32_t lds_byte_addr,
                                              uint32_t row_stride_elems,
                                              uint32_t tensor_rows) {
  u32x4 g0;
  i32x8 g1;
  i32x4 gz;
  gz[0] = 0; gz[1] = 0; gz[2] = 0; gz[3] = 0;

  const uint64_t ga = (uint64_t)(uintptr_t)gsrc;

  // group0: count=1 (valid user D#), lds_addr, global_addr[56:0], type=2
  g0[0] = 1u;
  g0[1] = lds_byte_addr;
  g0[2] = (uint32_t)ga;
  g0[3] = ((uint32_t)(ga >> 32) & 0x01FFFFFFu) | (2u << 30);

  // group1: data_size=2 (4B). tensor_dim0 = row_stride_elems (full row),
  // tensor_dim1 = tensor_rows, tile_dim0 = TILE_W, tile_dim1 = CHUNK,
  // tile_dim2 = 0 (2D), tensor_dim0_stride = row_stride_elems.
  g1[0] = (int)(2u << 16);
  g1[1] = (int)((row_stride_elems & 0xFFFFu) << 16);
  g1[2] = (int)(((row_stride_elems >> 16) & 0xFFFFu) |
                ((tensor_rows & 0xFFFFu) << 16));
  g1[3] = (int)(((tensor_rows >> 16) & 0xFFFFu) | ((uint32_t)TILE_W << 16));
  g1[4] = (int)CHUNK;
  g1[5] = (int)row_stride_elems;
  g1[6] = 0;
  g1[7] = 0;

#if defined(__clang_major__) && (__clang_major__ >= 23)
  i32x8 gz8;
  gz8[0]=0; gz8[1]=0; gz8[2]=0; gz8[3]=0; gz8[4]=0; gz8[5]=0; gz8[6]=0; gz8[7]=0;
  __builtin_amdgcn_tensor_load_to_lds(g0, g1, gz, gz, gz8, 0);
#else
  __builtin_amdgcn_tensor_load_to_lds(g0, g1, gz, gz, 0);
#endif
}

__global__ void __launch_bounds__(BLOCK)
qrnn_forgetmult_tdm(const float* __restrict__ f, const float* __restrict__ z,
                    const float* __restrict__ ig, const float* __restrict__ h0,
                    float* __restrict__ out, int T, int BH) {
  // [wave][buf(2)][arr(3)][CHUNK*TILE_W] = 8*2*3*512 floats = 96 KB LDS
  __shared__ float smem[WAVES * 2 * 3 * CHUNK * TILE_W];

  const int tid  = threadIdx.x;
  const int lane = tid & 31;
  const int wave = __builtin_amdgcn_readfirstlane(tid >> 5);  // scalar wave id

  const int col0 = blockIdx.x * BLOCK + wave * TILE_W;  // first channel of wave
  const int ch   = col0 + lane;                         // this lane's channel

  const uint32_t tile_elems = CHUNK * TILE_W;
  const uint32_t tile_bytes = tile_elems * 4u;
  const uint32_t lds_base   = (uint32_t)(uintptr_t)(void*)smem;
  const uint32_t wbase      = lds_base + (uint32_t)wave * 2u * 3u * tile_bytes;

  const int nchunks = T / CHUNK;

  // Prologue: stage chunk 0 into buffer 0 (3 TDM ops -> TENSORcnt=3).
  {
    const size_t off = (size_t)col0;  // t0 = 0
    tdm_load_tile(f  + off, wbase + 0u * tile_bytes, (uint32_t)BH, (uint32_t)T);
    tdm_load_tile(z  + off, wbase + 1u * tile_bytes, (uint32_t)BH, (uint32_t)T);
    tdm_load_tile(ig + off, wbase + 2u * tile_bytes, (uint32_t)BH, (uint32_t)T);
  }

  float h = h0[ch];

  for (int c = 0; c < nchunks; ++c) {
    if (c + 1 < nchunks) {
      // Prefetch next chunk into the other buffer, then wait until only those
      // 3 are outstanding -> current chunk's DMA is complete (in-order).
      const uint32_t nxt = wbase + (uint32_t)((c + 1) & 1) * 3u * tile_bytes;
      const size_t off = (size_t)(c + 1) * CHUNK * (size_t)BH + (size_t)col0;
      tdm_load_tile(f  + off, nxt + 0u * tile_bytes, (uint32_t)BH, (uint32_t)T);
      tdm_load_tile(z  + off, nxt + 1u * tile_bytes, (uint32_t)BH, (uint32_t)T);
      tdm_load_tile(ig + off, nxt + 2u * tile_bytes, (uint32_t)BH, (uint32_t)T);
      __builtin_amdgcn_s_wait_tensorcnt(3);
    } else {
      __builtin_amdgcn_s_wait_tensorcnt(0);
    }
    // TDM wrote LDS behind the compiler's back; fence the optimizer.
    asm volatile("" ::: "memory");

    float* buf = &smem[(size_t)((wave * 2 + (c & 1)) * 3) * tile_elems];
    const float* lf = buf;
    const float* lz = buf + tile_elems;
    const float* li = buf + 2 * tile_elems;

    size_t ob = (size_t)c * CHUNK * (size_t)BH + (size_t)ch;
#pragma unroll
    for (int t = 0; t < CHUNK; ++t) {
      const float fv = lf[t * TILE_W + lane];
      const float zv = lz[t * TILE_W + lane];
      const float iv = li[t * TILE_W + lane];
      h = fmaf(fv, h, iv * zv);            // h = i*z + f*h_prev
      out[ob + (size_t)t * BH] = h;        // coalesced 128B per wave
    }
  }
}

extern "C" void kernel_launch(void* const* d_in, const int* in_sizes, int n_in,
                              void* d_out, int out_size, void* d_ws, size_t ws_size,
                              hipStream_t stream) {
  const float* f  = (const float*)d_in[0];
  const float* z  = (const float*)d_in[1];
  const float* ig = (const float*)d_in[2];
  const float* h0 = (const float*)d_in[3];
  float* out = (float*)d_out;

  const int BH = in_sizes[3];       // B*H (hidden_init elements) = 8192
  const int T  = in_sizes[0] / BH;  // 4096

  const int blocks = BH / BLOCK;    // 32 blocks of 256 threads (8 waves)
  qrnn_forgetmult_tdm<<<blocks, BLOCK, 0, stream>>>(f, z, ig, h0, out, T, BH);
}